// PagedAttentionOp_13812614824014
// MI455X (gfx1250) — compile-verified
//
#include <hip/hip_runtime.h>
#include <hip/hip_bf16.h>

// Paged-KV decode attention for MI455X (gfx1250), wave32 + WMMA f32 16x16x4
// + CDNA5 async DMA (global_load_async_to_lds_b128) for V-tile staging.
//
// query:       [B=32, 1, N=32, D=128] f32
// key_pages:   [n_kv=8, 4096, T=32, D=128] f32
// value_pages: [n_kv=8, 4096, T=32, D=128] f32
// lengths:     [B] i32     page_map: [B, P=64] i32
// out:         [B, 1, N, D] f32
//
// One workgroup per (kv_head, batch). 8 waves split the 2048 KV tokens into
// 16-token tiles; each wave runs an online-softmax (flash) accumulation with
// V_WMMA_F32_16X16X4_F32 for both Q.K^T and P.V. V tiles are staged into a
// per-wave double-buffered LDS region by the async-to-LDS DMA (ASYNCcnt),
// overlapping the next tile's V fetch with the current tile's QK + softmax.

typedef __attribute__((ext_vector_type(2))) float v2f;
typedef __attribute__((ext_vector_type(8))) float v8f;

#define B_      32
#define N_      32
#define D_      128
#define NKV     8
#define G_      4
#define P_      64
#define T_      32
#define NPAGES  4096
#define WAVES   8
#define NEG_INF (-1.0e10f)
#define VSTRIDE 136   // V LDS row stride (floats): 128 + 8 pad -> conflict-free ds reads

__global__ __launch_bounds__(256) void paged_attn_wmma_kernel(
    const float* __restrict__ query,
    const float* __restrict__ key_pages,
    const float* __restrict__ value_pages,
    const int*   __restrict__ lengths,
    const int*   __restrict__ page_map,
    float*       __restrict__ out)
{
    const int kh   = blockIdx.x;      // kv head
    const int b    = blockIdx.y;      // batch slot
    const int tid  = threadIdx.x;
    const int wave = tid >> 5;
    const int lane = tid & 31;
    const int nlo  = lane & 15;       // N index (token / d-within-tile / q-row)
    const int h    = lane >> 4;       // half-wave select (K-pair select in A/B layout)

    __shared__ float vbuf[WAVES][2][16 * VSTRIDE];   // async-DMA'd V tiles (double buffered)
    __shared__ float lds_p[WAVES][16 * 16];          // per-wave P transpose scratch
    __shared__ float lds_acc[WAVES][G_][D_];         // per-wave partial outputs
    __shared__ float lds_m[WAVES][G_];
    __shared__ float lds_l[WAVES][G_];

    const int len = lengths[b];

    // ---- Load Q into WMMA A-matrix layout (16x4 f32 chunks over D). ----
    // Chunk c covers d = 4c..4c+3; lane holds A[M=nlo][4c+2h], A[M=nlo][4c+2h+1].
    // Rows M >= G are zeroed (their outputs are discarded; zero rows also keep
    // their softmax numerics bounded).
    const float* qbase = query + ((size_t)b * N_ + (size_t)kh * G_) * D_;
    v2f qa[32];
#pragma unroll
    for (int c = 0; c < 32; ++c) {
        if (nlo < G_) {
            qa[c] = *(const v2f*)(qbase + (size_t)nlo * D_ + 4 * c + 2 * h);
        } else {
            qa[c] = (v2f){0.f, 0.f};
        }
    }

    // ---- Online-softmax state (per C-layout row: VGPR j -> row j + 8h). ----
    float m_s[8], l_s[8];
    v8f   acc[8];                      // 8 D-tiles of 16 -> D=128
#pragma unroll
    for (int j = 0; j < 8; ++j) { m_s[j] = -1.0e30f; l_s[j] = 0.f; }
#pragma unroll
    for (int dt = 0; dt < 8; ++dt) {
        v8f z = {0.f, 0.f, 0.f, 0.f, 0.f, 0.f, 0.f, 0.f};
        acc[dt] = z;
    }

    const size_t head_off = (size_t)kh * NPAGES * T_ * D_;
    const float* kbase = key_pages + head_off;
    const float* vbase = value_pages + head_off;
    const int*   pmap  = page_map + (size_t)b * P_;

    // Issue the 16-row async DMA of one V tile into vbuf[wave][buf].
    // Row i (512B) is moved by one global_load_async_to_lds_b128: each lane
    // copies 16B.  LDS VDST operand = per-lane LDS byte offset (low 32 bits of
    // the generic LDS pointer, per the gfx1250 LDS aperture mapping).
    auto issue_v_dma = [&](const float* vt, int buf) {
        float* dst0 = &vbuf[wave][buf][0];
#pragma unroll
        for (int i = 0; i < 16; ++i) {
            unsigned long long g = (unsigned long long)(vt + (size_t)i * D_ + lane * 4);
            unsigned           l = (unsigned)(size_t)(dst0 + i * VSTRIDE + lane * 4);
            asm volatile("global_load_async_to_lds_b128 %0, %1, off"
                         :: "v"(l), "v"(g) : "memory");
        }
    };
    auto tile_kv_off = [&](int t) -> size_t {
        return ((size_t)pmap[t >> 1] * T_ + (size_t)((t & 1) * 16)) * D_;
    };

    // ---- Stream KV tiles (16 tokens each; a tile never crosses a page). ----
    int par = 0;
    if (wave * 16 < len) issue_v_dma(vbase + tile_kv_off(wave), 0);   // prologue

    for (int t = wave; t * 16 < len; t += WAVES) {
        const float* kt = kbase + tile_kv_off(t);

        const int  tn       = t + WAVES;
        const bool has_next = (tn * 16 < len);
        if (has_next) {
            // Kick next tile's V DMA into the other buffer; prefetch next K rows.
            const size_t noff = tile_kv_off(tn);
            issue_v_dma(vbase + noff, par ^ 1);
            __builtin_prefetch(kbase + noff + (size_t)nlo * D_, 0, 0);
        }

        // ---- scores[16x16] = Q[16xD] . K^T[Dx16], 32 chunks of K-depth 4. ----
        // B layout: lane holds B[K=4c+2h][N=nlo], B[K=4c+2h+1][N=nlo]
        //         = key[token=nlo][d=4c+2h], key[token=nlo][d=4c+2h+1].
        v8f sc = {0.f, 0.f, 0.f, 0.f, 0.f, 0.f, 0.f, 0.f};
        const float* krow = kt + (size_t)nlo * D_ + 2 * h;
#pragma unroll
        for (int c = 0; c < 32; ++c) {
            v2f kb = *(const v2f*)(krow + 4 * c);
            sc = __builtin_amdgcn_wmma_f32_16x16x4_f32(
                     false, qa[c], false, kb, (short)0, sc, false, false);
        }

        // ---- Masked online softmax (row j lives in sc[j]; tokens = lanes). ----
        const int  tok   = t * 16 + nlo;
        const bool valid = tok < len;
        float p_s[8];
#pragma unroll
        for (int j = 0; j < 8; ++j) {
            float s  = valid ? sc[j] : NEG_INF;
            float tm = s;
            tm = fmaxf(tm, __shfl_xor(tm, 1, 32));
            tm = fmaxf(tm, __shfl_xor(tm, 2, 32));
            tm = fmaxf(tm, __shfl_xor(tm, 4, 32));
            tm = fmaxf(tm, __shfl_xor(tm, 8, 32));
            const float mn   = fmaxf(m_s[j], tm);
            const float corr = __expf(m_s[j] - mn);
            const float p    = valid ? __expf(s - mn) : 0.f;
            float ps = p;
            ps += __shfl_xor(ps, 1, 32);
            ps += __shfl_xor(ps, 2, 32);
            ps += __shfl_xor(ps, 4, 32);
            ps += __shfl_xor(ps, 8, 32);
            l_s[j] = l_s[j] * corr + ps;
            m_s[j] = mn;
            p_s[j] = p;
#pragma unroll
            for (int dt = 0; dt < 8; ++dt) acc[dt][j] *= corr;
        }

        // ---- Transpose P (C layout -> A layout) through per-wave LDS. ----
        // LDS ops are in-order within a wave; no barrier needed.
#pragma unroll
        for (int j = 0; j < 8; ++j)
            lds_p[wave][(j + 8 * h) * 16 + nlo] = p_s[j];

        // ---- Release the current V buffer: 16 async ops per tile complete
        // in order, so <=16 outstanding means this tile's DMA is done while
        // the next tile's 16 ops may still be in flight. ----
        if (has_next) asm volatile("s_wait_asynccnt 0x10" ::: "memory");
        else          asm volatile("s_wait_asynccnt 0x0"  ::: "memory");

        // ---- out[16xD] += P[16x16] . V[16xD], 4 chunks of K-depth 4. ----
        // B operand read from the DMA'd LDS tile (stride-136 rows: lanes 0-15
        // and 16-31 land on disjoint 16-bank groups -> conflict-free).
        const float* vtile = &vbuf[wave][par][0];
#pragma unroll
        for (int c2 = 0; c2 < 4; ++c2) {
            v2f pa = *(const v2f*)&lds_p[wave][nlo * 16 + 4 * c2 + 2 * h];
            const float* vrow = vtile + (4 * c2 + 2 * h) * VSTRIDE + nlo;
#pragma unroll
            for (int dt = 0; dt < 8; ++dt) {
                v2f vb;
                vb[0] = vrow[dt * 16];            // V[token=4c2+2h  ][d=dt*16+nlo]
                vb[1] = vrow[dt * 16 + VSTRIDE];  // V[token=4c2+2h+1][d=dt*16+nlo]
                acc[dt] = __builtin_amdgcn_wmma_f32_16x16x4_f32(
                              false, pa, false, vb, (short)0, acc[dt], false, false);
            }
        }
        par ^= 1;
    }

    // ---- Publish per-wave partials (only rows 0..G-1 are real). ----
    if (h == 0) {
#pragma unroll
        for (int j = 0; j < G_; ++j) {
#pragma unroll
            for (int dt = 0; dt < 8; ++dt)
                lds_acc[wave][j][dt * 16 + nlo] = acc[dt][j];
            if (nlo == 0) { lds_m[wave][j] = m_s[j]; lds_l[wave][j] = l_s[j]; }
        }
    }
    __syncthreads();

    // ---- Flash merge across the 8 waves: 4 rows x 128 d = 512 outputs. ----
#pragma unroll
    for (int i = 0; i < 2; ++i) {
        const int idx = tid + i * 256;
        const int row = idx >> 7;        // 0..3
        const int d   = idx & 127;
        float M = -1.0e30f;
#pragma unroll
        for (int w = 0; w < WAVES; ++w) M = fmaxf(M, lds_m[w][row]);
        float num = 0.f, den = 0.f;
#pragma unroll
        for (int w = 0; w < WAVES; ++w) {
            const float e = __expf(lds_m[w][row] - M);
            num += e * lds_acc[w][row][d];
            den += e * lds_l[w][row];
        }
        out[((size_t)b * N_ + (size_t)(kh * G_ + row)) * D_ + d] = num / den;
    }
}

extern "C" void kernel_launch(void* const* d_in, const int* in_sizes, int n_in,
                              void* d_out, int out_size, void* d_ws, size_t ws_size,
                              hipStream_t stream) {
    (void)in_sizes; (void)n_in; (void)out_size; (void)d_ws; (void)ws_size;
    const float* query       = (const float*)d_in[0];
    const float* key_pages   = (const float*)d_in[1];
    const float* value_pages = (const float*)d_in[2];
    const int*   lengths     = (const int*)d_in[3];
    const int*   page_map    = (const int*)d_in[4];
    float*       out         = (float*)d_out;

    dim3 grid(NKV, B_);
    dim3 block(256);
    paged_attn_wmma_kernel<<<grid, block, 0, stream>>>(
        query, key_pages, value_pages, lengths, page_map, out);
}